// GATConvNet_35716948034093
// MI455X (gfx1250) — compile-verified
//
#include <hip/hip_runtime.h>

typedef __attribute__((ext_vector_type(2))) float v2f;
typedef __attribute__((ext_vector_type(8))) float v8f;
typedef __attribute__((ext_vector_type(4))) int i32x4;

#define NEG_SLOPE 0.2f
#define EPSV 1e-16f

// ---------- order-preserving float<->uint encoding for atomic max ----------
__device__ __forceinline__ unsigned fenc(float f) {
  unsigned b = __float_as_uint(f);
  return (b & 0x80000000u) ? ~b : (b | 0x80000000u);
}
__device__ __forceinline__ float fdec(unsigned u) {
  return (u & 0x80000000u) ? __uint_as_float(u & 0x7fffffffu)
                           : __uint_as_float(~u);
}
// fenc(-inf) == 0x007FFFFF
#define ENC_NEG_INF 0x007FFFFFu

__device__ __forceinline__ float leaky(float x) {
  return x > 0.f ? x : NEG_SLOPE * x;
}

#define AS_STRIDE 132  // even (b64-aligned pairs), 132%64==4 -> conflict-free
                       // column reads, 132*4%16==0 -> b128 LDS writes aligned

// =====================================================================
// Dual GEMM via fp32 WMMA: Hout = X@W ; Bout = X@S^T + cb + sb
// X:[N,128], W:[128,128] row-major (k,c), S:[128,128] row-major (c,k)
// grid.x = ceil(N/128), block = 256 (8 waves, wave32)
// B matrices staged in LDS pre-swizzled into WMMA lane order:
//   float2 {B[k][c], B[k+1][c]} at idx (kg*128+c)*2 + (khalf>>1)
// so each lane's operand is one aligned ds_load_b64, wave reads 256
// contiguous bytes (bank-conflict-free), no repacking movs.
// A tile staged via gfx1250 async global->LDS B128 (full tiles).
// =====================================================================
__global__ __launch_bounds__(256)
void gemm_dual_wmma(const float* __restrict__ X, const float* __restrict__ W,
                    const float* __restrict__ S, const float* __restrict__ cb,
                    const float* __restrict__ sb, float* __restrict__ Hout,
                    float* __restrict__ Bout, int N)
{
  __shared__ float  AS[128 * AS_STRIDE];  // ~67.6KB
  __shared__ float2 BwS[32 * 128 * 2];    // 64KB, WMMA-swizzled W
  __shared__ float2 BsS[32 * 128 * 2];    // 64KB, WMMA-swizzled S^T

  const int tid = threadIdx.x;
  const int rowBase = blockIdx.x * 128;

  // ---- stage B matrices (swizzled) ----
  for (int i = tid; i < 32 * 128 * 2; i += 256) {
    const int p  = i & 1;          // k-half select (0 -> k0,k0+1 ; 1 -> k0+2,k0+3)
    const int c  = (i >> 1) & 127; // column
    const int kg = i >> 8;         // k-group (K/4)
    const int k  = kg * 4 + p * 2;
    BwS[i] = make_float2(W[(size_t)k * 128 + c], W[(size_t)(k + 1) * 128 + c]);
    // skip weight used transposed: B'[k][c] = S[c][k] -> contiguous float2
    BsS[i] = *reinterpret_cast<const float2*>(S + (size_t)c * 128 + k);
  }

  // ---- stage A tile ----
  bool asyncA = false;
#if __has_builtin(__builtin_amdgcn_global_load_async_to_lds_b128)
  if (rowBase + 128 <= N) {
    typedef __attribute__((address_space(1))) i32x4 gi32x4;
    typedef __attribute__((address_space(3))) i32x4 li32x4;
    for (int i = tid; i < 128 * 32; i += 256) {
      const int r  = i >> 5;
      const int c4 = (i & 31) << 2;
      __builtin_amdgcn_global_load_async_to_lds_b128(
          (gi32x4*)(X + (size_t)(rowBase + r) * 128 + c4),
          (li32x4*)(&AS[r * AS_STRIDE + c4]), 0, 0);
    }
    asyncA = true;
  }
#endif
  if (!asyncA) {
    for (int i = tid; i < 128 * 32; i += 256) {
      const int r  = i >> 5;
      const int c4 = (i & 31) << 2;
      const int gr = rowBase + r;
      float4 x4 = make_float4(0.f, 0.f, 0.f, 0.f);
      if (gr < N)
        x4 = *reinterpret_cast<const float4*>(X + (size_t)gr * 128 + c4);
      AS[r * AS_STRIDE + c4 + 0] = x4.x;
      AS[r * AS_STRIDE + c4 + 1] = x4.y;
      AS[r * AS_STRIDE + c4 + 2] = x4.z;
      AS[r * AS_STRIDE + c4 + 3] = x4.w;
    }
  }
#if __has_builtin(__builtin_amdgcn_global_load_async_to_lds_b128)
  asm volatile("s_wait_asynccnt 0" ::: "memory");
#endif
  __syncthreads();

  const int wave  = tid >> 5;
  const int lane  = tid & 31;
  const int lrow  = lane & 15;          // M (for A) / N (for B)
  const int khalf = (lane >> 4) << 1;   // 0 or 2 (K sub-offset per ISA layout)
  const int wr    = wave * 16;          // this wave's row strip

  const v8f vzero = {};
  v8f accH[8], accS[8];
#pragma unroll
  for (int j = 0; j < 8; ++j) { accH[j] = vzero; accS[j] = vzero; }

  const int arow = (wr + lrow) * AS_STRIDE + khalf;
  const int bofs = ((lane & 15) << 1) + (lane >> 4);

  for (int kg = 0; kg < 32; ++kg) {
    const float2 av = *reinterpret_cast<const float2*>(&AS[arow + kg * 4]);
    v2f a; a.x = av.x; a.y = av.y;
    const int bbase = kg * 256 + bofs;
#pragma unroll
    for (int j = 0; j < 8; ++j) {
      const float2 bwv = BwS[bbase + j * 32];
      const float2 bsv = BsS[bbase + j * 32];
      v2f bw; bw.x = bwv.x; bw.y = bwv.y;
      v2f bs; bs.x = bsv.x; bs.y = bsv.y;
      accH[j] = __builtin_amdgcn_wmma_f32_16x16x4_f32(
          false, a, false, bw, (short)0, accH[j], false, false);
      accS[j] = __builtin_amdgcn_wmma_f32_16x16x4_f32(
          false, a, false, bs, (short)0, accS[j], false, false);
    }
  }

  // C/D layout: VGPR v, lanes0-15 -> M=v ; lanes16-31 -> M=v+8 ; N=lane&15
  const int mofs = (lane >> 4) << 3;
  const int ncol = lane & 15;
#pragma unroll
  for (int j = 0; j < 8; ++j) {
    const int col = j * 16 + ncol;
    const float bias = cb[col] + sb[col];
#pragma unroll
    for (int v = 0; v < 8; ++v) {
      const int grow = rowBase + wr + mofs + v;
      if (grow < N) {
        Hout[(size_t)grow * 128 + col] = accH[j][v];
        Bout[(size_t)grow * 128 + col] = accS[j][v] + bias;
      }
    }
  }
}

// =====================================================================
// Per-node attention logits: a_s[n,h] = sum_c H[n,h*32+c]*asrc[h,c]
// =====================================================================
__global__ void att_logits_kernel(const float* __restrict__ Hf,
                                  const float* __restrict__ asrc,
                                  const float* __restrict__ adst,
                                  float* __restrict__ As_, float* __restrict__ Ad_,
                                  int N)
{
  const int idx = blockIdx.x * blockDim.x + threadIdx.x;
  if (idx >= N * 4) return;
  const int n = idx >> 2, h = idx & 3;
  const float* row = Hf + (size_t)n * 128 + h * 32;
  const float* ws = asrc + h * 32;
  const float* wd = adst + h * 32;
  float s = 0.f, d = 0.f;
#pragma unroll
  for (int c = 0; c < 32; ++c) {
    const float v = row[c];
    s = fmaf(v, ws[c], s);
    d = fmaf(v, wd[c], d);
  }
  As_[idx] = s;
  Ad_[idx] = d;
}

// =====================================================================
// Softmax state init
// =====================================================================
__global__ void init_softmax_kernel(unsigned* __restrict__ mEnc,
                                    float* __restrict__ sSum, long long tot)
{
  const long long i = blockIdx.x * (long long)blockDim.x + threadIdx.x;
  if (i < tot) { mEnc[i] = ENC_NEG_INF; sSum[i] = 0.f; }
}

// =====================================================================
// Edge pass 1: per-dst running max of leaky_relu(a_s[src]+a_d[dst])
// =====================================================================
__global__ void edge_max_kernel(const int* __restrict__ e0, const int* __restrict__ e1,
                                const float* __restrict__ As_, const float* __restrict__ Ad_,
                                unsigned* __restrict__ mEnc, int E, int N, int Hh)
{
  const long long idx = blockIdx.x * (long long)blockDim.x + threadIdx.x;
  const long long tot = (long long)(E + N) * Hh;
  if (idx >= tot) return;
  const int h = (int)(idx % Hh);
  const long long e = idx / Hh;
  const int src = (e < E) ? e0[e] : (int)(e - E);
  const int dst = (e < E) ? e1[e] : (int)(e - E);
  const float v = leaky(As_[(size_t)src * Hh + h] + Ad_[(size_t)dst * Hh + h]);
  atomicMax(&mEnc[(size_t)dst * Hh + h], fenc(v));
}

// =====================================================================
// Edge pass 2: exp(e - max) -> store per-edge, accumulate per-dst sum
// =====================================================================
__global__ void edge_expsum_kernel(const int* __restrict__ e0, const int* __restrict__ e1,
                                   const float* __restrict__ As_, const float* __restrict__ Ad_,
                                   const unsigned* __restrict__ mEnc,
                                   float* __restrict__ sSum, float* __restrict__ eA,
                                   int E, int N, int Hh)
{
  const long long idx = blockIdx.x * (long long)blockDim.x + threadIdx.x;
  const long long tot = (long long)(E + N) * Hh;
  if (idx >= tot) return;
  const int h = (int)(idx % Hh);
  const long long e = idx / Hh;
  const int src = (e < E) ? e0[e] : (int)(e - E);
  const int dst = (e < E) ? e1[e] : (int)(e - E);
  const float v = leaky(As_[(size_t)src * Hh + h] + Ad_[(size_t)dst * Hh + h]);
  const float m = fdec(mEnc[(size_t)dst * Hh + h]);
  const float ex = __expf(v - m);
  eA[idx] = ex;
  atomicAdd(&sSum[(size_t)dst * Hh + h], ex);
}

// =====================================================================
// Edge pass 3 (layers 0/1): out[dst,col] += h[src,col] * alpha(e, col>>5)
// one thread per (edge, col) -> fully coalesced gather + f32 atomics
// =====================================================================
__global__ void edge_scatter128_kernel(const int* __restrict__ e0, const int* __restrict__ e1,
                                       const float* __restrict__ Hf,
                                       const float* __restrict__ eA,
                                       const float* __restrict__ sSum,
                                       float* __restrict__ out, int E, int N)
{
  const long long idx = blockIdx.x * (long long)blockDim.x + threadIdx.x;
  const long long tot = (long long)(E + N) * 128;
  if (idx >= tot) return;
  const int col = (int)(idx & 127);
  const long long e = idx >> 7;
  const int h = col >> 5;
  const int src = (e < E) ? e0[e] : (int)(e - E);
  const int dst = (e < E) ? e1[e] : (int)(e - E);
  const float alpha = eA[e * 4 + h] / (sSum[(size_t)dst * 4 + h] + EPSV);
  atomicAdd(&out[(size_t)dst * 128 + col], Hf[(size_t)src * 128 + col] * alpha);
}

// Edge pass 3 for layer 2 (H=1, C=1)
__global__ void edge_scatter1_kernel(const int* __restrict__ e0, const int* __restrict__ e1,
                                     const float* __restrict__ H2,
                                     const float* __restrict__ eA,
                                     const float* __restrict__ sSum,
                                     float* __restrict__ out, int E, int N)
{
  const long long e = blockIdx.x * (long long)blockDim.x + threadIdx.x;
  if (e >= (long long)(E + N)) return;
  const int src = (e < E) ? e0[e] : (int)(e - E);
  const int dst = (e < E) ? e1[e] : (int)(e - E);
  const float alpha = eA[e] / (sSum[dst] + EPSV);
  atomicAdd(&out[dst], H2[src] * alpha);
}

__global__ void relu_kernel(float* __restrict__ a, long long tot)
{
  const long long i = blockIdx.x * (long long)blockDim.x + threadIdx.x;
  if (i < tot) a[i] = fmaxf(a[i], 0.f);
}

__global__ void sigmoid_kernel(const float* __restrict__ a, float* __restrict__ o, int N)
{
  const int i = blockIdx.x * blockDim.x + threadIdx.x;
  if (i < N) o[i] = 1.f / (1.f + __expf(-a[i]));
}

// =====================================================================
// Layer 2 per-node: wave-per-node dual dot over 128 features
// =====================================================================
__global__ __launch_bounds__(256)
void l2_node_kernel(const float* __restrict__ h,
                    const float* __restrict__ c2w, const float* __restrict__ s2w,
                    const float* __restrict__ asrc, const float* __restrict__ adst,
                    const float* __restrict__ c2b, const float* __restrict__ s2b,
                    float* __restrict__ H2, float* __restrict__ As_,
                    float* __restrict__ Ad_, float* __restrict__ B2, int N)
{
  const int wave = threadIdx.x >> 5;
  const int lane = threadIdx.x & 31;
  const int n = blockIdx.x * 8 + wave;
  if (n >= N) return;
  float d1 = 0.f, d2 = 0.f;
#pragma unroll
  for (int k = 0; k < 4; ++k) {
    const int c = lane + k * 32;
    const float v = h[(size_t)n * 128 + c];
    d1 = fmaf(v, c2w[c], d1);
    d2 = fmaf(v, s2w[c], d2);
  }
#pragma unroll
  for (int m = 16; m >= 1; m >>= 1) {
    d1 += __shfl_xor(d1, m, 32);
    d2 += __shfl_xor(d2, m, 32);
  }
  if (lane == 0) {
    H2[n]  = d1;
    As_[n] = d1 * asrc[0];
    Ad_[n] = d1 * adst[0];
    B2[n]  = d2 + c2b[0] + s2b[0];
  }
}

// =====================================================================
// Host-side orchestration
// =====================================================================
static inline unsigned cdiv_u(long long a, long long b) {
  return (unsigned)((a + b - 1) / b);
}

extern "C" void kernel_launch(void* const* d_in, const int* in_sizes, int n_in,
                              void* d_out, int out_size, void* d_ws, size_t ws_size,
                              hipStream_t stream)
{
  (void)n_in; (void)out_size; (void)ws_size;
  const float* x     = (const float*)d_in[0];
  const int*   ei    = (const int*)d_in[1];
  const float* c0_w  = (const float*)d_in[3];
  const float* c0_as = (const float*)d_in[4];
  const float* c0_ad = (const float*)d_in[5];
  const float* c0_b  = (const float*)d_in[6];
  const float* s0_w  = (const float*)d_in[7];
  const float* s0_b  = (const float*)d_in[8];
  const float* c1_w  = (const float*)d_in[9];
  const float* c1_as = (const float*)d_in[10];
  const float* c1_ad = (const float*)d_in[11];
  const float* c1_b  = (const float*)d_in[12];
  const float* s1_w  = (const float*)d_in[13];
  const float* s1_b  = (const float*)d_in[14];
  const float* c2_w  = (const float*)d_in[15];
  const float* c2_as = (const float*)d_in[16];
  const float* c2_ad = (const float*)d_in[17];
  const float* c2_b  = (const float*)d_in[18];
  const float* s2_w  = (const float*)d_in[19];
  const float* s2_b  = (const float*)d_in[20];

  const int N = in_sizes[0] / 128;
  const int E = in_sizes[1] / 2;
  const int* e0 = ei;
  const int* e1 = ei + E;

  // Workspace layout (floats)
  float* ws = (float*)d_ws;
  const size_t NN = (size_t)N;
  float*    Hbuf = ws;                      // N*128
  float*    Bbuf = Hbuf + NN * 128;         // N*128
  float*    Cbuf = Bbuf + NN * 128;         // N*128
  float*    asb  = Cbuf + NN * 128;         // N*4
  float*    adb  = asb + NN * 4;            // N*4
  float*    sSum = adb + NN * 4;            // N*4
  unsigned* mEnc = (unsigned*)(sSum + NN * 4); // N*4
  float*    eA   = (float*)(mEnc + NN * 4); // (E+N)*4
  float*    H2   = eA + (size_t)(E + N) * 4; // N
  float*    B2   = H2 + NN;                 // N

  const long long EN   = (long long)E + N;
  const long long tot4 = EN * 4;

  // ---------------- Layer 0 ----------------
  gemm_dual_wmma<<<cdiv_u(N, 128), 256, 0, stream>>>(x, c0_w, s0_w, c0_b, s0_b,
                                                     Hbuf, Bbuf, N);
  att_logits_kernel<<<cdiv_u((long long)N * 4, 256), 256, 0, stream>>>(
      Hbuf, c0_as, c0_ad, asb, adb, N);
  init_softmax_kernel<<<cdiv_u((long long)N * 4, 256), 256, 0, stream>>>(
      mEnc, sSum, (long long)N * 4);
  edge_max_kernel<<<cdiv_u(tot4, 256), 256, 0, stream>>>(e0, e1, asb, adb, mEnc, E, N, 4);
  edge_expsum_kernel<<<cdiv_u(tot4, 256), 256, 0, stream>>>(e0, e1, asb, adb, mEnc,
                                                            sSum, eA, E, N, 4);
  edge_scatter128_kernel<<<cdiv_u(EN * 128, 256), 256, 0, stream>>>(
      e0, e1, Hbuf, eA, sSum, Bbuf, E, N);
  relu_kernel<<<cdiv_u((long long)N * 128, 256), 256, 0, stream>>>(Bbuf, (long long)N * 128);

  // ---------------- Layer 1 ----------------
  gemm_dual_wmma<<<cdiv_u(N, 128), 256, 0, stream>>>(Bbuf, c1_w, s1_w, c1_b, s1_b,
                                                     Hbuf, Cbuf, N);
  att_logits_kernel<<<cdiv_u((long long)N * 4, 256), 256, 0, stream>>>(
      Hbuf, c1_as, c1_ad, asb, adb, N);
  init_softmax_kernel<<<cdiv_u((long long)N * 4, 256), 256, 0, stream>>>(
      mEnc, sSum, (long long)N * 4);
  edge_max_kernel<<<cdiv_u(tot4, 256), 256, 0, stream>>>(e0, e1, asb, adb, mEnc, E, N, 4);
  edge_expsum_kernel<<<cdiv_u(tot4, 256), 256, 0, stream>>>(e0, e1, asb, adb, mEnc,
                                                            sSum, eA, E, N, 4);
  edge_scatter128_kernel<<<cdiv_u(EN * 128, 256), 256, 0, stream>>>(
      e0, e1, Hbuf, eA, sSum, Cbuf, E, N);
  relu_kernel<<<cdiv_u((long long)N * 128, 256), 256, 0, stream>>>(Cbuf, (long long)N * 128);

  // ---------------- Layer 2 (heads=1, out=1) ----------------
  l2_node_kernel<<<cdiv_u(N, 8), 256, 0, stream>>>(Cbuf, c2_w, s2_w, c2_as, c2_ad,
                                                   c2_b, s2_b, H2, asb, adb, B2, N);
  init_softmax_kernel<<<cdiv_u(N, 256), 256, 0, stream>>>(mEnc, sSum, (long long)N);
  edge_max_kernel<<<cdiv_u(EN, 256), 256, 0, stream>>>(e0, e1, asb, adb, mEnc, E, N, 1);
  edge_expsum_kernel<<<cdiv_u(EN, 256), 256, 0, stream>>>(e0, e1, asb, adb, mEnc,
                                                          sSum, eA, E, N, 1);
  edge_scatter1_kernel<<<cdiv_u(EN, 256), 256, 0, stream>>>(e0, e1, H2, eA, sSum, B2, E, N);
  sigmoid_kernel<<<cdiv_u(N, 256), 256, 0, stream>>>(B2, (float*)d_out, N);
}